// GroupwiseCostVolume3D_74509092651569
// MI455X (gfx1250) — compile-verified
//
#include <hip/hip_runtime.h>

// GroupwiseCostVolume3D for MI455X (gfx1250, wave32).
// out[n,g,d,h,x] = (1/16) * sum_{c<16} L[n,g*16+c,h,x] * R[n,g*16+c,h,x-d], x>=d, else 0
// Band-limited Gram tiles D = L^T R via V_WMMA_F32_16X16X4_F32; staging via
// GLOBAL_LOAD_ASYNC_TO_LDS (ASYNCcnt-tracked) when the toolchain exposes the builtin.

typedef __attribute__((ext_vector_type(2))) float v2f;
typedef __attribute__((ext_vector_type(8))) float v8f;

#define NGRP 8
#define CTOT 128
#define CPG 16
#define HH 120
#define WW 160
#define DMAX 48
// x-major LDS: row = one x position, 16 channels + 4 pad floats.
// Lane addresses stride 20 floats: l*20 mod 64 hits all multiples of 4, and the
// two half-waves (K offset 0 vs 2) fill the gaps -> all 64 banks, conflict-free.
#define LDSS 20

#if __has_builtin(__builtin_amdgcn_global_load_async_to_lds_b32) && \
    __has_builtin(__builtin_amdgcn_s_wait_asynccnt)
#define USE_ASYNC_LDS 1
typedef __attribute__((address_space(1))) int* gptr_i32;
typedef __attribute__((address_space(3))) int* lptr_i32;
#else
#define USE_ASYNC_LDS 0
#endif

__global__ __launch_bounds__(256) void gcv_wmma_kernel(const float* __restrict__ L,
                                                       const float* __restrict__ R,
                                                       float* __restrict__ out) {
    __shared__ float Ls[WW * LDSS];
    __shared__ float Rs[WW * LDSS];

    const int slice = blockIdx.x;       // (n*8 + g)*120 + h
    const int h     = slice % HH;
    const int ng    = slice / HH;       // n*8 + g
    const int g     = ng % NGRP;
    const int n     = ng / NGRP;

    const int tid = threadIdx.x;

    // ---- stage this slice's 16-channel L/R rows into LDS, transposed to x-major ----
    // Global reads coalesced in x; LDS dst = x*20 + cc.
    const size_t in_base = (((size_t)n * CTOT + (size_t)g * CPG) * HH + h) * WW;
    for (int idx = tid; idx < CPG * WW; idx += 256) {
        const int cc = idx / WW;
        const int x  = idx - cc * WW;
        const size_t off = in_base + (size_t)cc * (HH * WW) + x;
        const int dst = x * LDSS + cc;
#if USE_ASYNC_LDS
        __builtin_amdgcn_global_load_async_to_lds_b32(
            (gptr_i32)(L + off), (lptr_i32)(Ls + dst), 0, 0);
        __builtin_amdgcn_global_load_async_to_lds_b32(
            (gptr_i32)(R + off), (lptr_i32)(Rs + dst), 0, 0);
#else
        Ls[dst] = L[off];
        Rs[dst] = R[off];
#endif
    }
#if USE_ASYNC_LDS
    __builtin_amdgcn_s_wait_asynccnt(0);  // own async copies landed in LDS
#endif
    __syncthreads();                      // everyone's copies landed

    // out flat index: ((ng*48 + d)*120 + h)*160 + x
    const size_t out_slice = (size_t)ng * (DMAX * HH * WW) + (size_t)h * WW;

    // ---- zero padding region x < d (harness poisons d_out) ----
    for (int idx = tid; idx < DMAX * DMAX; idx += 256) {
        const int d = idx / DMAX;
        const int x = idx - d * DMAX;
        if (x < d) out[out_slice + (size_t)d * (HH * WW) + x] = 0.0f;
    }

    // ---- band tiles of D = L^T R via v_wmma_f32_16x16x4_f32 ----
    const int wave = tid >> 5;   // 8 wave32 per block
    const int lane = tid & 31;
    const int mrow = lane & 15;  // M (for A), N (for B/D) within half-wave
    const int hi   = lane >> 4;  // half-wave: K offset 0/2 (A,B), M offset 0/8 (D)

    int t = 0;
    for (int xi = 0; xi < WW / 16; ++xi) {
        const int jmax = (xi < 3) ? xi : 3;       // y-tile yi = xi - j must be >= 0
        for (int j = 0; j <= jmax; ++j, ++t) {
            if ((t & 7) != wave) continue;        // wave-uniform: EXEC stays all-ones
            const int x0 = xi * 16;
            const int y0 = (xi - j) * 16;

            // Contiguous per-lane fragments (one ds_load_b64 each).
            const float* arow = &Ls[(x0 + mrow) * LDSS + hi * 2];
            const float* brow = &Rs[(y0 + mrow) * LDSS + hi * 2];

            v8f acc = {};
#pragma unroll
            for (int kk = 0; kk < 4; ++kk) {
                // ISA 32-bit A 16x4: lanes0-15 {K=0 (v0), K=1 (v1)}, lanes16-31 {K=2,K=3};
                // B 4x16 striped identically across half-waves.
                const v2f a = *(const v2f*)(arow + kk * 4);
                const v2f b = *(const v2f*)(brow + kk * 4);
                // 8 args: (neg_a, A, neg_b, B, c_mod, C, reuse_a, reuse_b)
                acc = __builtin_amdgcn_wmma_f32_16x16x4_f32(
                    false, a, false, b, (short)0, acc, false, false);
            }

            // D layout: VGPR i -> M = i + 8*hi, N = mrow. d = x - y = 16j + M - N.
#pragma unroll
            for (int i = 0; i < 8; ++i) {
                const int m = i + hi * 8;
                const int d = j * 16 + m - mrow;
                if (d >= 0 && d < DMAX) {
                    const int x = x0 + m;
                    out[out_slice + (size_t)d * (HH * WW) + x] = acc[i] * 0.0625f;
                }
            }
        }
    }
}

extern "C" void kernel_launch(void* const* d_in, const int* in_sizes, int n_in,
                              void* d_out, int out_size, void* d_ws, size_t ws_size,
                              hipStream_t stream) {
    const float* L = (const float*)d_in[0];
    const float* R = (const float*)d_in[1];
    float* out = (float*)d_out;
    // one workgroup (8 wave32) per (n, g, h) slice
    dim3 grid(8 * NGRP * HH);
    gcv_wmma_kernel<<<grid, 256, 0, stream>>>(L, R, out);
}